// GNN_26998164423369
// MI455X (gfx1250) — compile-verified
//
#include <hip/hip_runtime.h>

typedef __attribute__((ext_vector_type(2))) float v2f;
typedef __attribute__((ext_vector_type(8))) float v8f;

#define DD 128
#define LDS_STRIDE 132   // 128 + 4 pad -> bank = (4n + koff) % 64, conflict-free frag reads

// ---------------------------------------------------------------------------
// zero fill (float4 granularity, grid-stride)
// ---------------------------------------------------------------------------
__global__ void zero_f4(float4* __restrict__ p, long n4) {
    long i = (long)blockIdx.x * blockDim.x + threadIdx.x;
    long stride = (long)gridDim.x * blockDim.x;
    float4 z = make_float4(0.f, 0.f, 0.f, 0.f);
    for (; i < n4; i += stride) p[i] = z;
}

// ---------------------------------------------------------------------------
// degree: one thread per edge
// ---------------------------------------------------------------------------
__global__ void degree_kernel(const int* __restrict__ dst, float* __restrict__ deg, int E) {
    int e = blockIdx.x * blockDim.x + threadIdx.x;
    if (e < E) unsafeAtomicAdd(&deg[dst[e]], 1.0f);
}

// ---------------------------------------------------------------------------
// scatter-add: one wave per edge; lane handles a float4 chunk (coalesced 512B)
// ---------------------------------------------------------------------------
__global__ void scatter_add(const float* __restrict__ X,
                            const int* __restrict__ src,
                            const int* __restrict__ dst,
                            float* __restrict__ agg, int E) {
    int gw   = (blockIdx.x * blockDim.x + threadIdx.x) >> 5;
    int lane = threadIdx.x & 31;
    if (gw >= E) return;
    int s = src[gw];
    int d = dst[gw];
    const float4 v = *(const float4*)(X + (size_t)s * DD + lane * 4);
    float* o = agg + (size_t)d * DD + lane * 4;
    unsafeAtomicAdd(o + 0, v.x);
    unsafeAtomicAdd(o + 1, v.y);
    unsafeAtomicAdd(o + 2, v.z);
    unsafeAtomicAdd(o + 3, v.w);
}

// ---------------------------------------------------------------------------
// Fused SAGE layer GEMM:  Out = act( (Agg * invdeg) @ Wl + bias + X @ Wr )
// 256 threads = 8 waves; wave w computes rows [ (blk*8+w)*16 , +16 ) x all 128 cols
// f32 WMMA 16x16x4, both GEMM terms chained into the same accumulators.
// ---------------------------------------------------------------------------
__launch_bounds__(256, 1)
__global__ void sage_fused_gemm(const float* __restrict__ Agg,
                                const float* __restrict__ Deg,
                                const float* __restrict__ X,
                                const float* __restrict__ Wl,
                                const float* __restrict__ Bias,
                                const float* __restrict__ Wr,
                                float* __restrict__ Out,
                                int N, int doRelu) {
    // W staged transposed+padded: sW[n*132 + k] = W[k*128 + n]
    __shared__ __align__(16) float sWl[DD * LDS_STRIDE];
    __shared__ __align__(16) float sWr[DD * LDS_STRIDE];

    const int tid = threadIdx.x;
    for (int i = tid; i < DD * DD; i += 256) {
        int k = i >> 7;       // W row (K)
        int n = i & 127;      // W col (N)
        sWl[n * LDS_STRIDE + k] = Wl[i];
        sWr[n * LDS_STRIDE + k] = Wr[i];
    }
    __syncthreads();

    const int wave   = tid >> 5;
    const int lane   = tid & 31;
    const int laneHi = lane >> 4;      // 0: lanes 0-15, 1: lanes 16-31
    const int lane16 = lane & 15;
    const int rowBase = (blockIdx.x * 8 + wave) * 16;
    if (rowBase >= N) return;          // whole-wave exit (after the only barrier)

    const int row  = rowBase + lane16; // A-frag row for this lane (both halves)
    const float dg   = Deg[row];
    const float invd = 1.0f / fmaxf(dg, 1.0f);

    // A-frag layout: v0 holds K=k0 (lanes 0-15) / K=k0+2 (lanes 16-31), v1 = +1
    const float* aggRow = Agg + (size_t)row * DD + 2 * laneHi;
    const float* xRow   = X   + (size_t)row * DD + 2 * laneHi;

    v8f acc[8] = {};   // 8 accumulators: 16x16 tiles covering N = 0..127

    #pragma unroll 4
    for (int k0 = 0; k0 < DD; k0 += 4) {
        float2 t1 = *(const float2*)(aggRow + k0);
        float2 t2 = *(const float2*)(xRow + k0);
        v2f a1 = { t1.x * invd, t1.y * invd };   // mean = agg * invdeg (fused)
        v2f a2 = { t2.x, t2.y };
        const int kk = k0 + 2 * laneHi;          // B-frag K base for this lane half
        #pragma unroll
        for (int n = 0; n < 8; ++n) {
            const int nIdx = (n * 16 + lane16) * LDS_STRIDE + kk;
            float2 tl = *(const float2*)&sWl[nIdx];
            float2 tr = *(const float2*)&sWr[nIdx];
            v2f bl = { tl.x, tl.y };
            v2f br = { tr.x, tr.y };
            acc[n] = __builtin_amdgcn_wmma_f32_16x16x4_f32(
                false, a1, false, bl, (short)0, acc[n], false, false);
            acc[n] = __builtin_amdgcn_wmma_f32_16x16x4_f32(
                false, a2, false, br, (short)0, acc[n], false, false);
        }
    }

    // C/D layout: VGPR r, lanes 0-15 -> M=r, lanes 16-31 -> M=r+8; N = lane16
    #pragma unroll
    for (int n = 0; n < 8; ++n) {
        const int col = n * 16 + lane16;
        const float bn = Bias[col];
        #pragma unroll
        for (int r = 0; r < 8; ++r) {
            const int rowOut = rowBase + r + 8 * laneHi;
            float v = acc[n][r] + bn;
            if (doRelu) v = fmaxf(v, 0.0f);
            Out[(size_t)rowOut * DD + col] = v;
        }
    }
}

// ---------------------------------------------------------------------------
extern "C" void kernel_launch(void* const* d_in, const int* in_sizes, int n_in,
                              void* d_out, int out_size, void* d_ws, size_t ws_size,
                              hipStream_t stream) {
    const float* x   = (const float*)d_in[0];
    const int*   ei  = (const int*)d_in[1];   // [2, E] int
    const float* W1l = (const float*)d_in[2];
    const float* b1  = (const float*)d_in[3];
    const float* W1r = (const float*)d_in[4];
    const float* W2l = (const float*)d_in[5];
    const float* b2  = (const float*)d_in[6];
    const float* W2r = (const float*)d_in[7];

    const int N = in_sizes[0] / DD;       // 50000
    const int E = in_sizes[1] / 2;        // 640000
    const int* src = ei;
    const int* dst = ei + E;

    float* agg = (float*)d_ws;                     // N*128 f32
    float* h   = agg + (size_t)N * DD;             // N*128 f32
    float* deg = h   + (size_t)N * DD;             // N     f32

    const long nAgg4 = (long)N * DD / 4;
    const long nDeg4 = (long)N / 4;                // 50000 % 4 == 0

    const int zeroBlocks = 2048;
    const int edgeThreadBlocks = (E + 255) / 256;
    const int edgeWaveBlocks   = (E + 7) / 8;      // 8 waves (edges) per 256-thr block
    const int gemmBlocks       = (N / 16 + 7) / 8; // 8 row-tiles per block

    // ---- setup: degree (once) + zero agg ----
    zero_f4<<<zeroBlocks, 256, 0, stream>>>((float4*)agg, nAgg4);
    zero_f4<<<64, 256, 0, stream>>>((float4*)deg, nDeg4);
    degree_kernel<<<edgeThreadBlocks, 256, 0, stream>>>(dst, deg, E);

    // ---- layer 1 ----
    scatter_add<<<edgeWaveBlocks, 256, 0, stream>>>(x, src, dst, agg, E);
    sage_fused_gemm<<<gemmBlocks, 256, 0, stream>>>(agg, deg, x, W1l, b1, W1r, h, N, 1);

    // ---- layer 2 ----
    zero_f4<<<zeroBlocks, 256, 0, stream>>>((float4*)agg, nAgg4);
    scatter_add<<<edgeWaveBlocks, 256, 0, stream>>>(h, src, dst, agg, E);
    sage_fused_gemm<<<gemmBlocks, 256, 0, stream>>>(agg, deg, h, W2l, b2, W2r,
                                                    (float*)d_out, N, 0);
}